// VQEModel_67912022884830
// MI455X (gfx1250) — compile-verified
//
#include <hip/hip_runtime.h>

typedef __attribute__((ext_vector_type(2))) float v2f;
typedef __attribute__((ext_vector_type(8))) float v8f;

#define NQ 13
#define DD 8192            // 2^13
#define KSPLIT 32          // k-chunks per row tile
#define ROWTILES (DD / 16) // 512
#define KCHUNK (DD / KSPLIT)

// d_ws float layout:
//   [0   .. 224)  wr table [14][16]
//   [224 .. 448)  wi table [14][16]
//   [512 .. 512 + ROWTILES*KSPLIT)  per-block partial sums

// ---------------------------------------------------------------------------
// Stage 1: analytic single-layer product-state weights.
// a = e^{-i phi/2} cos(theta/2);  b = -i e^{+i phi/2} sin(theta/2)
// c_p = a^(N-p) b^p ;  wr+i*wi = conj(c_p) * c_q
// ---------------------------------------------------------------------------
__global__ void vqe_weights_kernel(const float* __restrict__ theta,
                                   const float* __restrict__ phi,
                                   float* __restrict__ ws) {
    if (threadIdx.x != 0) return;
    float th = theta[0], ph = phi[0];
    float ca = __cosf(th * 0.5f), sa = __sinf(th * 0.5f);
    float cp = __cosf(ph * 0.5f), sp = __sinf(ph * 0.5f);
    // a = ca*(cp - i sp) ; b = -i * sa * (cp + i sp) = sa*(sp - i cp)
    float ar = ca * cp, ai = -ca * sp;
    float br = sa * sp, bi = -sa * cp;

    float cr[NQ + 1], ci[NQ + 1];
    for (int p = 0; p <= NQ; ++p) {
        float xr = 1.0f, xi = 0.0f;
        for (int t = 0; t < NQ - p; ++t) {
            float nr = xr * ar - xi * ai;
            xi = xr * ai + xi * ar;
            xr = nr;
        }
        for (int t = 0; t < p; ++t) {
            float nr = xr * br - xi * bi;
            xi = xr * bi + xi * br;
            xr = nr;
        }
        cr[p] = xr; ci[p] = xi;
    }
    for (int p = 0; p <= NQ; ++p) {
        for (int q = 0; q < 16; ++q) {
            float wr = 0.0f, wi = 0.0f;
            if (q <= NQ) {
                wr = cr[p] * cr[q] + ci[p] * ci[q];   // Re(conj(c_p) c_q)
                wi = cr[p] * ci[q] - ci[p] * cr[q];   // Im(conj(c_p) c_q)
            }
            ws[p * 16 + q]       = wr;
            ws[224 + p * 16 + q] = wi;
        }
    }
}

// ---------------------------------------------------------------------------
// Stage 2: WMMA streaming kernel.  One wave per (16-row tile, 256-col chunk).
// C[r][q] += sum_k H[row_r, k] * [pop(k)==q]   via V_WMMA_F32_16X16X4_F32,
// then contract with wr/wi tables, wave-reduce, emit one partial per block.
// ---------------------------------------------------------------------------
__global__ void __launch_bounds__(32)
vqe_wmma_kernel(const float* __restrict__ Hr, const float* __restrict__ Hi,
                const float* __restrict__ wtab, float* __restrict__ partials) {
    const int lane    = threadIdx.x;       // 0..31 (wave32)
    const int rt      = blockIdx.x;        // row tile
    const int kc      = blockIdx.y;        // k chunk
    const int rowbase = rt * 16;
    const int row     = rowbase + (lane & 15);
    const int kstart  = kc * KCHUNK;
    const int koff    = (lane >> 4) * 2;   // upper half-wave owns K+2,K+3
    const int q       = lane & 15;

    v8f Cr = {}; v8f Ci = {};
    const float* pr = Hr + (size_t)row * DD + kstart + koff;
    const float* pi = Hi + (size_t)row * DD + kstart + koff;

    for (int k0 = kstart; k0 < kstart + KCHUNK; k0 += 4) {
        v2f a_r = *(const v2f*)pr;          // A tile: H[row, kb], H[row, kb+1]
        v2f a_i = *(const v2f*)pi;
        int kb = k0 + koff;
        v2f b;                              // B tile: indicator [pop(k)==q]
        b.x = (__popc(kb)     == q) ? 1.0f : 0.0f;
        b.y = (__popc(kb + 1) == q) ? 1.0f : 0.0f;
        Cr = __builtin_amdgcn_wmma_f32_16x16x4_f32(false, a_r, false, b,
                                                   (short)0, Cr, false, false);
        Ci = __builtin_amdgcn_wmma_f32_16x16x4_f32(false, a_i, false, b,
                                                   (short)0, Ci, false, false);
        pr += 4; pi += 4;
    }

    // Epilogue: C layout = VGPR v: lanes 0-15 -> M=v, lanes 16-31 -> M=v+8,
    // column N = lane & 15.
    float s = 0.0f;
    const int rhalf = (lane >> 4) * 8;
#pragma unroll
    for (int v = 0; v < 8; ++v) {
        int r = rowbase + rhalf + v;
        int p = __popc(r);
        float wr = wtab[p * 16 + q];
        float wi = wtab[224 + p * 16 + q];
        s += Cr[v] * wr - Ci[v] * wi;
    }
    // wave32 butterfly reduction
#pragma unroll
    for (int m = 16; m >= 1; m >>= 1) s += __shfl_xor(s, m, 32);
    if (lane == 0) partials[rt * KSPLIT + kc] = s;
}

// ---------------------------------------------------------------------------
// Stage 3: deterministic fixed-order final reduction.
// ---------------------------------------------------------------------------
__global__ void __launch_bounds__(256)
vqe_reduce_kernel(const float* __restrict__ partials, float* __restrict__ out,
                  int n) {
    __shared__ float sm[256];
    float s = 0.0f;
    for (int i = threadIdx.x; i < n; i += 256) s += partials[i];
    sm[threadIdx.x] = s;
    __syncthreads();
    for (int w = 128; w >= 1; w >>= 1) {
        if ((int)threadIdx.x < w) sm[threadIdx.x] += sm[threadIdx.x + w];
        __syncthreads();
    }
    if (threadIdx.x == 0) out[0] = sm[0];
}

extern "C" void kernel_launch(void* const* d_in, const int* in_sizes, int n_in,
                              void* d_out, int out_size, void* d_ws, size_t ws_size,
                              hipStream_t stream) {
    (void)in_sizes; (void)n_in; (void)out_size; (void)ws_size;
    const float* theta = (const float*)d_in[0];
    const float* phi   = (const float*)d_in[1];
    const float* Hr    = (const float*)d_in[2];
    const float* Hi    = (const float*)d_in[3];
    float* ws       = (float*)d_ws;
    float* partials = ws + 512;

    vqe_weights_kernel<<<1, 32, 0, stream>>>(theta, phi, ws);
    dim3 grid(ROWTILES, KSPLIT);
    vqe_wmma_kernel<<<grid, 32, 0, stream>>>(Hr, Hi, ws, partials);
    vqe_reduce_kernel<<<1, 256, 0, stream>>>(partials, (float*)d_out,
                                             ROWTILES * KSPLIT);
}